// RoutingPooling_24369644438074
// MI455X (gfx1250) — compile-verified
//
#include <hip/hip_runtime.h>
#include <cstdint>

// ---------------------------------------------------------------------------
// RoutingPooling: 2x2 stride-2 capsule routing pool, NUM_ROUTING=2, ATOMS=4.
// x: [32,64,64,256] f32  ->  out: [32,32,32,256] f32
// Per (b,i,j,group g of 4 channels): votes v0..v3 = the 4 window positions'
// float4; m = mean(v); l_k = <v_k,m>; r = softmax(l); pose = sum r_k v_k.
// Memory-bound (~168 MB traffic, ~7us floor @ 23.3 TB/s). Uses CDNA5 async
// global->LDS DMA path (global_load_async_to_lds_b128 + s_wait_asynccnt).
// ---------------------------------------------------------------------------

#if defined(__AMDGCN__)
#if __has_builtin(__builtin_amdgcn_global_load_async_to_lds_b128)
#define USE_ASYNC_LDS 1
#endif
#endif
#ifndef USE_ASYNC_LDS
#define USE_ASYNC_LDS 0
#endif

// Matches the builtin's parameter pointee type reported by clang:
// "__attribute__((__vector_size__(4 * sizeof(int)))) int"
typedef int v4i_t __attribute__((vector_size(4 * sizeof(int))));
typedef __attribute__((address_space(1))) v4i_t* gbl_v4i_p;
typedef __attribute__((address_space(3))) v4i_t* lds_v4i_p;

__device__ __forceinline__ float dot4f(const float4 a, const float4 b) {
    return a.x * b.x + a.y * b.y + a.z * b.z + a.w * b.w;
}

__global__ __launch_bounds__(256) void routing_pool_kernel(
    const float* __restrict__ x, float* __restrict__ out) {
    // 64 KB LDS: two input rows (h=2i, h=2i+1) over a 32-wide w window,
    // all 256 channels, stored as float4 chunks.
    __shared__ float4 sm[4096];

    const int t    = threadIdx.x;
    const unsigned tile = blockIdx.x;      // 2048 tiles
    const int half = tile & 1;             // which half of the output row (16 j's)
    const int i    = (tile >> 1) & 31;     // output row
    const int b    = tile >> 6;            // batch

    const float4* __restrict__ x4 = (const float4*)x;
    // float4 index of (b, h=2i, w=half*32, c=0); one w step = 64 float4s,
    // one h step = 64*64 = 4096 float4s.
    const int rowbase4 = ((b * 64 + 2 * i) * 64 + half * 32) * 64;

    // ---- Stage: 4096 float4 chunks (2 rows x 32 w x 64 chunks) -------------
#pragma unroll
    for (int u = 0; u < 16; ++u) {
        const int c = t + 256 * u;         // chunk id in [0,4096)
        const int r = c >> 11;             // 0 or 1: which h row
        const int q = c & 2047;            // chunk within row window
        const float4* src = x4 + rowbase4 + r * 4096 + q;
#if USE_ASYNC_LDS
        __builtin_amdgcn_global_load_async_to_lds_b128(
            (gbl_v4i_p)src,
            (lds_v4i_p)&sm[c],
            /*offset=*/0, /*cpol=*/0);
#else
        sm[c] = *src;
#endif
    }

#if USE_ASYNC_LDS
#if __has_builtin(__builtin_amdgcn_s_wait_asynccnt)
    __builtin_amdgcn_s_wait_asynccnt(0);
#else
    asm volatile("s_wait_asynccnt 0" ::: "memory");
#endif
#endif
    __syncthreads();

    // ---- Compute: 1024 output groups per block, 4 per thread ---------------
    float4* __restrict__ out4 = (float4*)out;
#pragma unroll
    for (int u = 0; u < 4; ++u) {
        const int o  = t + 256 * u;        // [0,1024)
        const int jj = o >> 6;             // local output col in [0,16)
        const int g  = o & 63;             // capsule group

        // window positions: (w=2jj, w=2jj+1) x (row0, row1); 64 chunks per w.
        const int w0 = (2 * jj) * 64 + g;
        const int w1 = (2 * jj + 1) * 64 + g;
        const float4 v0 = sm[w0];          // (h0,w0)
        const float4 v1 = sm[w1];          // (h0,w1)
        const float4 v2 = sm[2048 + w0];   // (h1,w0)
        const float4 v3 = sm[2048 + w1];   // (h1,w1)

        // iter 0: uniform routing -> m = mean of votes
        float4 m;
        m.x = 0.25f * (v0.x + v1.x + v2.x + v3.x);
        m.y = 0.25f * (v0.y + v1.y + v2.y + v3.y);
        m.z = 0.25f * (v0.z + v1.z + v2.z + v3.z);
        m.w = 0.25f * (v0.w + v1.w + v2.w + v3.w);

        // agreement logits
        const float l0 = dot4f(v0, m);
        const float l1 = dot4f(v1, m);
        const float l2 = dot4f(v2, m);
        const float l3 = dot4f(v3, m);

        // iter 1: softmax over the 4 window positions
        const float mm = fmaxf(fmaxf(l0, l1), fmaxf(l2, l3));
        const float e0 = __expf(l0 - mm);
        const float e1 = __expf(l1 - mm);
        const float e2 = __expf(l2 - mm);
        const float e3 = __expf(l3 - mm);
        const float inv = 1.0f / (e0 + e1 + e2 + e3);

        float4 pose;
        pose.x = (e0 * v0.x + e1 * v1.x + e2 * v2.x + e3 * v3.x) * inv;
        pose.y = (e0 * v0.y + e1 * v1.y + e2 * v2.y + e3 * v3.y) * inv;
        pose.z = (e0 * v0.z + e1 * v1.z + e2 * v2.z + e3 * v3.z) * inv;
        pose.w = (e0 * v0.w + e1 * v1.w + e2 * v2.w + e3 * v3.w) * inv;

        // out[b, i, j = half*16 + jj, 4g .. 4g+3]
        const int oidx = ((b * 32 + i) * 32 + half * 16 + jj) * 64 + g;
        out4[oidx] = pose;
    }
}

extern "C" void kernel_launch(void* const* d_in, const int* in_sizes, int n_in,
                              void* d_out, int out_size, void* d_ws, size_t ws_size,
                              hipStream_t stream) {
    (void)in_sizes; (void)n_in; (void)out_size; (void)d_ws; (void)ws_size;
    const float* x  = (const float*)d_in[0];
    float* out      = (float*)d_out;
    // 2048 blocks: (batch=32) x (output row=32) x (row half=2), 256 thr each.
    routing_pool_kernel<<<dim3(2048), dim3(256), 0, stream>>>(x, out);
}